// FP_Layer_5583457485367
// MI455X (gfx1250) — compile-verified
//
#include <hip/hip_runtime.h>
#include <hip/hip_bf16.h>

#define BB 16
#define NL 4096
#define NH 1024
#define CL 128
#define CH 256
#define IN_CH 384
#define OUT_CH 256
#define LDC 388          // 384 + 4 pad floats: conflict-free b64 LDS reads, 16B-aligned rows
#define NBLK (BB * (NL / 16))   // 4096 GEMM blocks -> BN partial count

typedef __attribute__((ext_vector_type(2))) float v2f;
typedef __attribute__((ext_vector_type(4))) float v4f;
typedef __attribute__((ext_vector_type(8))) float v8f;

// ---------------------------------------------------------------------------
// Kernel 1: tiled transpose feat_high (B, CH, NH) -> fht (B, NH, CH)
// so neighbor gathers become contiguous 256-float rows.
// ---------------------------------------------------------------------------
__global__ void transpose_fh_kernel(const float* __restrict__ fh,
                                    float* __restrict__ fht) {
    __shared__ float tile[32][33];
    const int b  = blockIdx.z;
    const int i0 = blockIdx.x * 32;   // NH dim
    const int c0 = blockIdx.y * 32;   // CH dim
    const int tx = threadIdx.x;       // 32
    const int ty = threadIdx.y;       // 8
    #pragma unroll
    for (int k = 0; k < 32; k += 8)
        tile[ty + k][tx] = fh[((size_t)b * CH + (c0 + ty + k)) * NH + i0 + tx];
    __syncthreads();
    #pragma unroll
    for (int k = 0; k < 32; k += 8)
        fht[((size_t)b * NH + (i0 + ty + k)) * CH + c0 + tx] = tile[tx][ty + k];
}

// ---------------------------------------------------------------------------
// Kernel 2: three_nn — top-3 nearest high points per low point + IDW weights.
// xyz_high (+ squared norms) staged in LDS (16 KB), broadcast reads.
// Matches reference metric (sq_l + sq_h - 2*inner, clipped) and tie order.
// ---------------------------------------------------------------------------
__global__ void three_nn_kernel(const float* __restrict__ xyz_low,
                                const float* __restrict__ xyz_high,
                                int* __restrict__ idx_out,
                                float* __restrict__ w_out) {
    __shared__ float4 sh[NH];
    const int b = blockIdx.y;
    const int t = threadIdx.x;  // 256
    const float* xh = xyz_high + (size_t)b * NH * 3;
    for (int i = t; i < NH; i += 256) {
        float x = xh[i * 3 + 0], y = xh[i * 3 + 1], z = xh[i * 3 + 2];
        sh[i] = make_float4(x, y, z, x * x + y * y + z * z);
    }
    __syncthreads();

    const int l = blockIdx.x * 256 + t;
    const float* xl = xyz_low + ((size_t)b * NL + l) * 3;
    const float px = xl[0], py = xl[1], pz = xl[2];
    const float sql = px * px + py * py + pz * pz;

    float d0 = 3.4e38f, d1 = 3.4e38f, d2 = 3.4e38f;
    int   i0 = 0, i1 = 0, i2 = 0;
    for (int h = 0; h < NH; ++h) {
        float4 p = sh[h];
        float inner = px * p.x + py * p.y + pz * p.z;
        float d = fmaxf(sql + p.w - 2.0f * inner, 0.0f);
        if (d < d0)       { d2 = d1; i2 = i1; d1 = d0; i1 = i0; d0 = d; i0 = h; }
        else if (d < d1)  { d2 = d1; i2 = i1; d1 = d;  i1 = h; }
        else if (d < d2)  { d2 = d;  i2 = h; }
    }
    float r0 = 1.0f / fmaxf(sqrtf(d0), 1e-8f);
    float r1 = 1.0f / fmaxf(sqrtf(d1), 1e-8f);
    float r2 = 1.0f / fmaxf(sqrtf(d2), 1e-8f);
    float inv = 1.0f / (r0 + r1 + r2);
    const int base = ((b * NL) + l) * 3;
    idx_out[base + 0] = i0; idx_out[base + 1] = i1; idx_out[base + 2] = i2;
    w_out[base + 0] = r0 * inv; w_out[base + 1] = r1 * inv; w_out[base + 2] = r2 * inv;
}

// ---------------------------------------------------------------------------
// Kernel 3: fused interpolate + concat + GEMM (y = W @ f_cat + b) via
// V_WMMA_F32_16X16X4_F32, plus per-block BN partial sums computed from the
// register-resident output tile (avoids a full 64 MB re-read for BN stats).
// Block = 512 threads (16 waves); each wave owns a 16(o)x16(n) tile.
// ---------------------------------------------------------------------------
__global__ void __launch_bounds__(512)
interp_gemm_kernel(const float* __restrict__ Wm,
                   const float* __restrict__ bias,
                   const float* __restrict__ fht,       // (B, NH, CH)
                   const float* __restrict__ feat_low,  // (B, CL, NL)
                   const int* __restrict__ idx,
                   const float* __restrict__ wgt,
                   float* __restrict__ y,               // (B, OUT_CH, NL)
                   float* __restrict__ psum,            // [OUT_CH][NBLK]
                   float* __restrict__ psq) {           // [OUT_CH][NBLK]
    __shared__ float F[16 * LDC];   // [n][c], c-major rows, 24.8 KB
    const int b  = blockIdx.z;
    const int n0 = blockIdx.x * 16;
    const int t  = threadIdx.x;

    // ---- Phase 1: interpolated features, c in [0, 256) ----
    {
        const int n = t >> 5;        // 0..15
        const int q = t & 31;        // float4-quad lane
        const int base = ((b * NL) + n0 + n) * 3;
        const int j0 = idx[base + 0], j1 = idx[base + 1], j2 = idx[base + 2];
        const float w0 = wgt[base + 0], w1 = wgt[base + 1], w2 = wgt[base + 2];
        const float* r0 = fht + ((size_t)b * NH + j0) * CH;
        const float* r1 = fht + ((size_t)b * NH + j1) * CH;
        const float* r2 = fht + ((size_t)b * NH + j2) * CH;
        #pragma unroll
        for (int j = 0; j < 2; ++j) {
            const int c = (q + 32 * j) * 4;           // 0..255, 16B aligned
            v4f a0 = *(const v4f*)(r0 + c);
            v4f a1 = *(const v4f*)(r1 + c);
            v4f a2 = *(const v4f*)(r2 + c);
            v4f f  = a0 * w0 + a1 * w1 + a2 * w2;
            *(v4f*)(&F[n * LDC + c]) = f;
        }
    }
    // ---- Phase 2: copy feat_low, c in [256, 384), coalesced over n ----
    {
        const int n = t & 15;
        const int c = t >> 4;        // 0..31
        #pragma unroll
        for (int j = 0; j < 4; ++j) {
            const int cc = c + 32 * j;                // 0..127
            F[n * LDC + 256 + cc] =
                feat_low[((size_t)b * CL + cc) * NL + n0 + n];
        }
    }
    __syncthreads();

    // ---- Phase 3: WMMA GEMM, one 16x16 tile per wave, K = 384 in steps of 4
    const int wave = t >> 5;         // 0..15 -> o-tile
    const int lane = t & 31;
    const int half = lane >> 4;      // K-pair select
    const int m    = lane & 15;      // A-matrix row / B-matrix col
    const int o0   = wave * 16;
    const float* Wrow = Wm + (size_t)(o0 + m) * IN_CH;
    const float* Frow = F + m * LDC;

    v8f acc = {};
    #pragma unroll 4
    for (int kk = 0; kk < IN_CH; kk += 4) {
        const int ko = kk + 2 * half;
        v2f a  = *(const v2f*)(Wrow + ko);     // A 16x4: lanes0-15 K0|K1, lanes16-31 K2|K3
        v2f bm = *(const v2f*)(Frow + ko);     // B 4x16: same K striping, N = lane&15
        acc = __builtin_amdgcn_wmma_f32_16x16x4_f32(
            /*neg_a=*/false, a, /*neg_b=*/false, bm,
            /*c_mod=*/(short)0, acc, /*reuse_a=*/false, /*reuse_b=*/false);
    }

    // ---- Epilogue: add conv bias, store y, and emit BN partial sums.
    // C/D layout: reg r, half h -> M = r + 8*h, N = lane&15.
    const int n = lane & 15;
    float vals[8];
    #pragma unroll
    for (int r = 0; r < 8; ++r) {
        const int o = o0 + r + 8 * half;
        float v = acc[r] + bias[o];
        vals[r] = v;
        y[((size_t)b * OUT_CH + o) * NL + n0 + n] = v;
    }
    // Butterfly over the 16 n's of each half (masks 1..8 stay within a half;
    // lanes 0-15 and 16-31 hold disjoint channel ranges). Fixed tree order ->
    // deterministic. Each block touches every channel exactly once, so the
    // partial needs no intra-block combine.
    const int bid = blockIdx.z * (NL / 16) + blockIdx.x;   // 0..NBLK-1
    #pragma unroll
    for (int r = 0; r < 8; ++r) {
        float s = vals[r];
        float q = vals[r] * vals[r];
        #pragma unroll
        for (int msk = 1; msk <= 8; msk <<= 1) {
            s += __shfl_xor(s, msk, 32);
            q += __shfl_xor(q, msk, 32);
        }
        if (n == 0) {
            const int o = o0 + r + 8 * half;
            psum[(size_t)o * NBLK + bid] = s;
            psq [(size_t)o * NBLK + bid] = q;
        }
    }
}

// ---------------------------------------------------------------------------
// Kernel 4: reduce BN partials per channel -> scale/shift (coalesced reads)
// ---------------------------------------------------------------------------
__global__ void bn_stats_kernel(const float* __restrict__ psum,
                                const float* __restrict__ psq,
                                const float* __restrict__ gamma,
                                const float* __restrict__ beta,
                                float* __restrict__ stats) {
    const int o = blockIdx.x;    // channel
    const int t = threadIdx.x;   // 256
    float s = 0.0f, q = 0.0f;
    for (int i = t; i < NBLK; i += 256) {
        s += psum[(size_t)o * NBLK + i];
        q += psq [(size_t)o * NBLK + i];
    }
    __shared__ float sh_s[256], sh_q[256];
    sh_s[t] = s; sh_q[t] = q;
    __syncthreads();
    for (int k = 128; k > 0; k >>= 1) {
        if (t < k) { sh_s[t] += sh_s[t + k]; sh_q[t] += sh_q[t + k]; }
        __syncthreads();
    }
    if (t == 0) {
        const float inv_n = 1.0f / (float)(BB * NL);
        float mean = sh_s[0] * inv_n;
        float var  = sh_q[0] * inv_n - mean * mean;
        float sc   = gamma[o] * rsqrtf(var + 1e-5f);
        stats[o]          = sc;
        stats[OUT_CH + o] = beta[o] - mean * sc;
    }
}

// ---------------------------------------------------------------------------
// Kernel 5: in-place normalize + ReLU, float4-vectorized.
// NL = 4096 -> a float4 never crosses a channel; o = (i4 >> 10) & 255.
// ---------------------------------------------------------------------------
__global__ void bn_apply_kernel(float* __restrict__ y,
                                const float* __restrict__ stats) {
    const size_t total4 = (size_t)BB * OUT_CH * NL / 4;
    const size_t stride = (size_t)gridDim.x * blockDim.x;
    v4f* y4 = (v4f*)y;
    for (size_t i = (size_t)blockIdx.x * blockDim.x + threadIdx.x;
         i < total4; i += stride) {
        const int o = (int)((i >> 10) & 255);
        const float sc = stats[o];
        const float sf = stats[OUT_CH + o];
        v4f v = y4[i];
        v4f r;
        #pragma unroll
        for (int k = 0; k < 4; ++k) r[k] = fmaxf(v[k] * sc + sf, 0.0f);
        y4[i] = r;
    }
}

// ---------------------------------------------------------------------------
extern "C" void kernel_launch(void* const* d_in, const int* in_sizes, int n_in,
                              void* d_out, int out_size, void* d_ws, size_t ws_size,
                              hipStream_t stream) {
    const float* xyz_low   = (const float*)d_in[0];
    const float* xyz_high  = (const float*)d_in[1];
    const float* feat_low  = (const float*)d_in[2];
    const float* feat_high = (const float*)d_in[3];
    const float* Wm        = (const float*)d_in[4];
    const float* bias      = (const float*)d_in[5];
    const float* gamma     = (const float*)d_in[6];
    const float* beta      = (const float*)d_in[7];
    float* y = (float*)d_out;   // (B, OUT_CH, NL); also serves as pre-BN buffer

    // Workspace layout (all chunks 256B-aligned):
    //   idx   : B*NL*3 int32        (768 KB)
    //   wgt   : B*NL*3 f32          (768 KB)
    //   stats : 2*OUT_CH f32        (2 KB)
    //   fht   : B*NH*CH f32         (16 MB)  transposed feat_high
    //   psum  : OUT_CH*NBLK f32     (4 MB)   BN partial sums
    //   psq   : OUT_CH*NBLK f32     (4 MB)   BN partial sums of squares
    char* ws = (char*)d_ws;
    const size_t idx_bytes  = (size_t)BB * NL * 3 * 4;
    const size_t fht_off    = 2 * idx_bytes + 2048;
    const size_t fht_bytes  = (size_t)BB * NH * CH * 4;
    const size_t psum_off   = fht_off + fht_bytes;
    const size_t psq_off    = psum_off + (size_t)OUT_CH * NBLK * 4;

    int*   idx   = (int*)ws;
    float* wgt   = (float*)(ws + idx_bytes);
    float* stats = (float*)(ws + 2 * idx_bytes);
    float* fht   = (float*)(ws + fht_off);
    float* psum  = (float*)(ws + psum_off);
    float* psq   = (float*)(ws + psq_off);

    transpose_fh_kernel<<<dim3(NH / 32, CH / 32, BB), dim3(32, 8), 0, stream>>>(
        feat_high, fht);
    three_nn_kernel<<<dim3(NL / 256, BB), 256, 0, stream>>>(
        xyz_low, xyz_high, idx, wgt);
    interp_gemm_kernel<<<dim3(NL / 16, 1, BB), 512, 0, stream>>>(
        Wm, bias, fht, feat_low, idx, wgt, y, psum, psq);
    bn_stats_kernel<<<OUT_CH, 256, 0, stream>>>(psum, psq, gamma, beta, stats);
    bn_apply_kernel<<<4096, 256, 0, stream>>>(y, stats);
}